// ConvLogicLayer_41223096107629
// MI455X (gfx1250) — compile-verified
//
#include <hip/hip_runtime.h>
#include <cmath>

// ---------------------------------------------------------------------------
// ConvLogicLayer for MI455X (gfx1250, wave32).
//
// Roofline: 256 MB f32 output write dominates (~11us @ 23.3 TB/s HBM).
// x (8 MB) lives in L2 (192 MB). Strategy:
//   * Stage 1 (WMMA): coef[c][p][j] = softmax(weights[c][p][:]) @ OP_COEFFS
//     -> exact f32 matmul (M=2048,K=16,N=4) via 4x chained
//        v_wmma_f32_16x16x4_f32 per wave (16 rows/wave, 128 waves).
//     B operand (OP_COEFFS columns) is built from compile-time constants with
//     v_cndmask selects -- no memory traffic / EXEC juggling between WMMAs.
//   * Stage 2: gather + fused logic-op evaluation; one thread = one pixel's
//     4 outputs, written as a single 128-bit NON-TEMPORAL store (write-once
//     stream, keep x hot in L2). selection/coef are wave-uniform -> s_loads.
// ---------------------------------------------------------------------------

typedef __attribute__((ext_vector_type(2))) float v2f;
typedef __attribute__((ext_vector_type(4))) float f4;
typedef __attribute__((ext_vector_type(8))) float v8f;

// OP_COEFFS[i][j], j = {k0, ka, kb, kab} -- compile-time only (folds to
// cndmask immediates; never materialized in memory).
constexpr float OPCC[16][4] = {
    {0.f, 0.f, 0.f, 0.f},  {0.f, 0.f, 0.f, 1.f},
    {0.f, 1.f, 0.f,-1.f},  {0.f, 1.f, 0.f, 0.f},
    {0.f, 0.f, 1.f,-1.f},  {0.f, 0.f, 1.f, 0.f},
    {0.f, 1.f, 1.f,-2.f},  {0.f, 1.f, 1.f,-1.f},
    {1.f,-1.f,-1.f, 1.f},  {1.f,-1.f,-1.f, 2.f},
    {1.f, 0.f,-1.f, 0.f},  {1.f, 0.f,-1.f, 1.f},
    {1.f,-1.f, 0.f, 0.f},  {1.f,-1.f, 0.f, 1.f},
    {1.f, 0.f, 0.f,-1.f},  {1.f, 0.f, 0.f, 0.f}};

// lane-select of OP_COEFFS row `K` at column n (0 for n >= 4): pure VALU.
__device__ __forceinline__ float opc_lane(int K, int n) {
    float r = 0.f;
    r = (n == 0) ? OPCC[K][0] : r;
    r = (n == 1) ? OPCC[K][1] : r;
    r = (n == 2) ? OPCC[K][2] : r;
    r = (n == 3) ? OPCC[K][3] : r;
    return r;
}

// ---------------------------------------------------------------------------
// Stage 1: one wave32 per 16 rows of the (2048 x 16) softmax-probability
// matrix. D(16x16) += A(16x4) x B(4x16) chained 4x over K.
// ISA 32-bit operand layouts (CDNA5 ISA §7.12.2):
//   A 16x4:  lanes 0-15 -> M=lane, {VGPR0,VGPR1} = K{0,1};
//            lanes 16-31 -> M=lane-16, K{2,3}.
//   B 4x16:  lane = N;   lanes 0-15 -> K{0,1}, lanes 16-31 -> K{2,3}.
//   C/D 16x16 f32: VGPR j: lanes 0-15 -> M=j, N=lane;
//                          lanes 16-31 -> M=j+8, N=lane-16.
// ---------------------------------------------------------------------------
__global__ __launch_bounds__(32) void coef_wmma_kernel(
    const float* __restrict__ weights,   // (2048, 16)
    float* __restrict__ coef)            // (2048, 4)
{
    const int lane     = threadIdx.x;          // 0..31
    const int row_base = blockIdx.x * 16;      // 128 blocks
    const int r        = lane & 15;            // A-row this lane feeds
    const bool hi      = lane >= 16;           // upper half holds K+2,K+3
    const int n        = lane & 15;            // B/N column

    // softmax over this row's 16 weights (lanes r and r+16 both need row r)
    const float* wr = weights + (row_base + r) * 16;
    float p[16];
    float m = wr[0];
#pragma unroll
    for (int i = 0; i < 16; ++i) { p[i] = wr[i]; m = fmaxf(m, p[i]); }
    float s = 0.f;
#pragma unroll
    for (int i = 0; i < 16; ++i) { p[i] = expf(p[i] - m); s += p[i]; }
    const float inv = 1.f / s;
#pragma unroll
    for (int i = 0; i < 16; ++i) p[i] *= inv;

    v8f d = {};   // C/D accumulator
#pragma unroll
    for (int t = 0; t < 4; ++t) {            // K chunks of 4
        v2f a, b;
        a[0] = hi ? p[4 * t + 2] : p[4 * t + 0];
        a[1] = hi ? p[4 * t + 3] : p[4 * t + 1];
        // B from immediates: lane<16 -> rows 4t,4t+1; lane>=16 -> 4t+2,4t+3
        b[0] = hi ? opc_lane(4 * t + 2, n) : opc_lane(4 * t + 0, n);
        b[1] = hi ? opc_lane(4 * t + 3, n) : opc_lane(4 * t + 1, n);
        // 8 args: (neg_a, A, neg_b, B, c_mod, C, reuse_a, reuse_b)
        d = __builtin_amdgcn_wmma_f32_16x16x4_f32(
                false, a, false, b, (short)0, d, false, false);
    }

    // scatter D: only N columns 0..3 are meaningful
    const int mrow = row_base + (hi ? 8 : 0);
    if (n < 4) {
#pragma unroll
        for (int j = 0; j < 8; ++j)
            coef[(mrow + j) * 4 + n] = d[j];
    }
}

// ---------------------------------------------------------------------------
// Stage 2: one block per (n, c_out) 32x32 tile; 256 threads x 4 pixels each.
// Per pixel: gather 8 taps (all in L2), evaluate 4 fused logic ops, emit one
// 128-bit non-temporal store (output is write-once, 256 MB total).
// ---------------------------------------------------------------------------
__global__ __launch_bounds__(256) void conv_logic_kernel(
    const float* __restrict__ x,     // (32, 64, 32, 32)
    const int*   __restrict__ sel,   // (512, 8)
    const float* __restrict__ coef,  // (512, 4, 4)
    float*       __restrict__ y)     // (32, 512, 32, 32, 4)
{
    const int bid = blockIdx.x;      // n*512 + c
    const int c   = bid & 511;

    // wave-uniform metadata -> scalar loads
    int ch[8], dy[8], dx[8];
#pragma unroll
    for (int i = 0; i < 8; ++i) {
        const int sv = sel[c * 8 + i];
        ch[i] = (sv >> 16) & 0xffff;
        dy[i] = ((sv >> 8) & 0xff) - 1;  // PAD_H = 1
        dx[i] = (sv & 0xff) - 1;         // PAD_W = 1
    }
    float k[16];
#pragma unroll
    for (int i = 0; i < 16; ++i) k[i] = coef[c * 16 + i];

    const float* xn = x + (size_t)(bid >> 9) * (64 * 32 * 32);
    f4* yb = (f4*)(y + (size_t)bid * (32 * 32 * 4));

    const int tid = threadIdx.x;
#pragma unroll
    for (int q = 0; q < 4; ++q) {
        const int pix = tid + q * 256;
        const int oy = pix >> 5;
        const int ox = pix & 31;
        f4 out;
#pragma unroll
        for (int p = 0; p < 4; ++p) {
            const int ia = 2 * p, ib = 2 * p + 1;
            const int ay = oy + dy[ia], ax = ox + dx[ia];
            const int by = oy + dy[ib], bx = ox + dx[ib];
            const float a = ((unsigned)ay < 32u && (unsigned)ax < 32u)
                ? xn[(ch[ia] << 10) + (ay << 5) + ax] : 0.f;
            const float b = ((unsigned)by < 32u && (unsigned)bx < 32u)
                ? xn[(ch[ib] << 10) + (by << 5) + bx] : 0.f;
            // k0 + ka*a + kb*b + kab*(a*b), fully fused
            out[p] = fmaf(k[4 * p + 3], a * b,
                      fmaf(k[4 * p + 2], b,
                       fmaf(k[4 * p + 1], a, k[4 * p])));
        }
        __builtin_nontemporal_store(out, yb + pix);   // 128-bit NT stream
    }
}

// ---------------------------------------------------------------------------
extern "C" void kernel_launch(void* const* d_in, const int* in_sizes, int n_in,
                              void* d_out, int out_size, void* d_ws, size_t ws_size,
                              hipStream_t stream) {
    const float* x       = (const float*)d_in[0];  // (32,64,32,32) f32
    const float* weights = (const float*)d_in[1];  // (512,4,16)    f32
    const int*   sel     = (const int*)d_in[2];    // (512,8)       i32
    float*       out     = (float*)d_out;          // (32,512,32,32,4) f32
    float*       coef    = (float*)d_ws;           // 2048*4 floats = 32 KB

    // Stage 1: 2048 rows / 16 rows-per-wave = 128 single-wave blocks
    coef_wmma_kernel<<<128, 32, 0, stream>>>(weights, coef);
    // Stage 2: 32*512 = 16384 tiles
    conv_logic_kernel<<<32 * 512, 256, 0, stream>>>(x, sel, coef, out);
}